// AttentionPooling2_84473416778478
// MI455X (gfx1250) — compile-verified
//
#include <hip/hip_runtime.h>
#include <hip/hip_bf16.h>
#include <math.h>

#define B_ 8
#define S_ 1024
#define N_ 128
#define D_ 256
#define LN_EPS 1e-5f
#define NEG_ (-10000.0f)

typedef __attribute__((ext_vector_type(2))) float v2f;
typedef __attribute__((ext_vector_type(8))) float v8f;

// ---------------------------------------------------------------------------
// Kernel 1: scorer MLP  -> scores (B*S)
// Block = 512 threads (16 waves), handles 16 consecutive tokens.
// Wave w computes H tile (16 tokens x 16 cols [16w,16w+16)) via
// v_wmma_f32_16x16x4_f32 over K=256, stages H in LDS, then wave w does the
// LayerNorm + erf-GELU + W2 dot for token w with wave32 reductions.
// ---------------------------------------------------------------------------
__global__ __launch_bounds__(512) void scorer_kernel(
    const float* __restrict__ doc, const float* __restrict__ W1,
    const float* __restrict__ b1, const float* __restrict__ gamma,
    const float* __restrict__ beta, const float* __restrict__ W2,
    const float* __restrict__ b2, float* __restrict__ scores)
{
    __shared__ float lds_h[16][D_ + 4];   // +4 pad to spread LDS banks

    const int tile = blockIdx.x;          // 0 .. B*S/16 - 1 (flat token tiles)
    const int tok0 = tile * 16;
    const int tid  = threadIdx.x;
    const int wave = tid >> 5;            // 0..15
    const int lane = tid & 31;
    const int lrow = lane & 15;           // M (A) / N (B,C) index
    const int lhi  = lane >> 4;           // 0: K={0,1}, 1: K={2,3}

    // ---- phase 1: 16x16 H tile per wave, fp32 WMMA, K = 0..255 step 4 ----
    const int ncol0 = wave * 16;
    const float* arow = doc + (size_t)(tok0 + lrow) * D_ + lhi * 2;
    v8f c = {0.f, 0.f, 0.f, 0.f, 0.f, 0.f, 0.f, 0.f};

    #pragma unroll 4
    for (int k0 = 0; k0 < D_; k0 += 4) {
        // A (16x4): lane holds (M=lrow, K=k0+2*lhi+{0,1})
        v2f a;
        a.x = arow[k0];
        a.y = arow[k0 + 1];
        // B (4x16): lane holds (K=k0+2*lhi+{0,1}, N=ncol0+lrow)
        const int kb = k0 + lhi * 2;
        v2f bf;
        bf.x = W1[(size_t)kb       * D_ + ncol0 + lrow];
        bf.y = W1[(size_t)(kb + 1) * D_ + ncol0 + lrow];
        c = __builtin_amdgcn_wmma_f32_16x16x4_f32(
                false, a, false, bf, (short)0, c, false, false);
    }

    // C layout: VGPR r, lane -> (M = r + 8*lhi, N = lrow); add b1 and stash.
    const int colw = ncol0 + lrow;
    const float bias = b1[colw];
    #pragma unroll
    for (int r = 0; r < 8; ++r) {
        lds_h[r + lhi * 8][colw] = c[r] + bias;
    }
    __syncthreads();

    // ---- phase 2: wave w -> token w : LayerNorm + GELU(erf) + dot(W2) ----
    {
        const int t = wave;
        float hv[8];
        float sum = 0.f, sumsq = 0.f;
        #pragma unroll
        for (int i = 0; i < 8; ++i) {
            float v = lds_h[t][lane + i * 32];
            hv[i] = v;
            sum += v;
            sumsq += v * v;
        }
        #pragma unroll
        for (int off = 16; off > 0; off >>= 1) {
            sum   += __shfl_xor(sum,   off, 32);
            sumsq += __shfl_xor(sumsq, off, 32);
        }
        const float mu  = sum * (1.f / D_);
        const float var = sumsq * (1.f / D_) - mu * mu;
        const float rs  = rsqrtf(var + LN_EPS);

        float acc = 0.f;
        #pragma unroll
        for (int i = 0; i < 8; ++i) {
            const int cidx = lane + i * 32;
            float hn = (hv[i] - mu) * rs * gamma[cidx] + beta[cidx];
            float g  = 0.5f * hn * (1.f + erff(hn * 0.7071067811865475f));
            acc += g * W2[cidx];
        }
        #pragma unroll
        for (int off = 16; off > 0; off >>= 1) acc += __shfl_xor(acc, off, 32);
        if (lane == 0) scores[tok0 + t] = acc + b2[0];
    }
}

// ---------------------------------------------------------------------------
// Kernel 2: masked softmax weights  P[b,n,s] = mapping * softmax_s(logits)
// One block (256 threads) per (b,n) row of S=1024.
// ---------------------------------------------------------------------------
__global__ __launch_bounds__(256) void softmax_kernel(
    const float* __restrict__ mapping, const float* __restrict__ scores,
    float* __restrict__ P)
{
    const int bn = blockIdx.x;            // b*N + n
    const int b  = bn / N_;
    const float* mrow = mapping + (size_t)bn * S_;
    const float* srow = scores  + (size_t)b * S_;
    float* prow = P + (size_t)bn * S_;
    const int tid = threadIdx.x;

    __shared__ float redmax[8];
    __shared__ float redsum[8];

    float mp[4], lg[4];
    float lmax = -3.4e38f;
    #pragma unroll
    for (int i = 0; i < 4; ++i) {
        const int s = tid + i * 256;
        const float m  = mrow[s];
        const float sc = srow[s];
        const float logit = m * sc + (1.f - m) * NEG_;
        mp[i] = m;
        lg[i] = logit;
        lmax = fmaxf(lmax, logit);
    }
    #pragma unroll
    for (int off = 16; off > 0; off >>= 1)
        lmax = fmaxf(lmax, __shfl_xor(lmax, off, 32));
    if ((tid & 31) == 0) redmax[tid >> 5] = lmax;
    __syncthreads();
    float bmax = redmax[0];
    #pragma unroll
    for (int w = 1; w < 8; ++w) bmax = fmaxf(bmax, redmax[w]);

    float e[4];
    float lsum = 0.f;
    #pragma unroll
    for (int i = 0; i < 4; ++i) {
        e[i] = __expf(lg[i] - bmax);
        lsum += e[i];
    }
    #pragma unroll
    for (int off = 16; off > 0; off >>= 1) lsum += __shfl_xor(lsum, off, 32);
    if ((tid & 31) == 0) redsum[tid >> 5] = lsum;
    __syncthreads();
    float denom = 0.f;
    #pragma unroll
    for (int w = 0; w < 8; ++w) denom += redsum[w];
    const float inv = 1.f / denom;

    #pragma unroll
    for (int i = 0; i < 4; ++i) {
        const int s = tid + i * 256;
        prow[s] = mp[i] * e[i] * inv;
    }
}

// ---------------------------------------------------------------------------
// Kernel 3: pooled = P (N x S) @ doc (S x D) per batch, fp32 WMMA.
// One 16x16 output tile per wave; 8 waves/block; 128 blocks total.
// ---------------------------------------------------------------------------
__global__ __launch_bounds__(256) void pool_kernel(
    const float* __restrict__ P, const float* __restrict__ doc,
    float* __restrict__ out)
{
    const int tid  = threadIdx.x;
    const int wave = tid >> 5;
    const int lane = tid & 31;
    const int lrow = lane & 15;
    const int lhi  = lane >> 4;

    const int tileId = blockIdx.x * 8 + wave;   // 0..1023
    const int b      = tileId >> 7;             // / (8*16)
    const int t      = tileId & 127;
    const int ntile  = t >> 4;                  // 0..7  (node tiles)
    const int dtile  = t & 15;                  // 0..15 (dim tiles)

    const float* Pr = P + ((size_t)b * N_ + ntile * 16 + lrow) * S_ + lhi * 2;
    const float* Dc = doc + (size_t)b * S_ * D_ + dtile * 16 + lrow;

    v8f c = {0.f, 0.f, 0.f, 0.f, 0.f, 0.f, 0.f, 0.f};
    #pragma unroll 4
    for (int k0 = 0; k0 < S_; k0 += 4) {
        v2f a;                                  // A: (M=node row, K)
        a.x = Pr[k0];
        a.y = Pr[k0 + 1];
        const int kb = k0 + lhi * 2;
        v2f bf;                                 // B: (K=seq, N=dim col)
        bf.x = Dc[(size_t)kb       * D_];
        bf.y = Dc[(size_t)(kb + 1) * D_];
        c = __builtin_amdgcn_wmma_f32_16x16x4_f32(
                false, a, false, bf, (short)0, c, false, false);
    }

    float* orow = out + ((size_t)b * N_ + ntile * 16) * D_ + dtile * 16 + lrow;
    #pragma unroll
    for (int r = 0; r < 8; ++r) {
        orow[(size_t)(r + lhi * 8) * D_] = c[r];
    }
}

// ---------------------------------------------------------------------------
extern "C" void kernel_launch(void* const* d_in, const int* in_sizes, int n_in,
                              void* d_out, int out_size, void* d_ws, size_t ws_size,
                              hipStream_t stream) {
    (void)in_sizes; (void)n_in; (void)out_size; (void)ws_size;

    const float* doc     = (const float*)d_in[0];  // (B,S,D)
    const float* mapping = (const float*)d_in[1];  // (B,N,S)
    /* d_in[2] = nodes_len (unused by forward) */
    const float* W1      = (const float*)d_in[3];  // (D,D)
    const float* b1      = (const float*)d_in[4];  // (D)
    const float* gamma   = (const float*)d_in[5];  // (D)
    const float* beta    = (const float*)d_in[6];  // (D)
    const float* W2      = (const float*)d_in[7];  // (D,1)
    const float* b2      = (const float*)d_in[8];  // (1)
    float* out           = (float*)d_out;          // (B,N,D)

    // workspace: scores (B*S floats) | P (B*N*S floats)
    float* scores = (float*)d_ws;
    float* Pbuf   = scores + (size_t)B_ * S_;

    scorer_kernel<<<(B_ * S_) / 16, 512, 0, stream>>>(
        doc, W1, b1, gamma, beta, W2, b2, scores);

    softmax_kernel<<<B_ * N_, 256, 0, stream>>>(mapping, scores, Pbuf);

    pool_kernel<<<(B_ * N_ / 16) * (D_ / 16) / 8, 256, 0, stream>>>(
        Pbuf, doc, out);
}